// GlobalLayer_44942537785492
// MI455X (gfx1250) — compile-verified
//
#include <hip/hip_runtime.h>
#include <math.h>

typedef __attribute__((ext_vector_type(2))) float v2f;
typedef __attribute__((ext_vector_type(8))) float v8f;

#define KFEAT 48
#define HID   16

// ---------------------------------------------------------------- zero ws
__global__ void zero_f4_kernel(float4* __restrict__ p, int n4) {
  int i = blockIdx.x * blockDim.x + threadIdx.x;
  if (i < n4) p[i] = make_float4(0.f, 0.f, 0.f, 0.f);
}

// ------------------------------------------------- scatter-add (gather+atomic)
// edge layout: [2, E] int64; row0 = src, row1 = dst. thread = (edge, 4-feat chunk)
__global__ void scatter_add_kernel(const float* __restrict__ x,
                                   const long long* __restrict__ edge,
                                   float* __restrict__ out, int E) {
  int tid = blockIdx.x * blockDim.x + threadIdx.x;
  int total = E * (KFEAT / 4);
  if (tid >= total) return;
  int e = tid / (KFEAT / 4);
  int q = tid - e * (KFEAT / 4);
  long long s = edge[e];
  long long d = edge[E + e];
  float4 v = *(const float4*)(x + s * (long long)KFEAT + q * 4);
  float* dp = out + d * (long long)KFEAT + q * 4;
  unsafeAtomicAdd(dp + 0, v.x);   // global_atomic_add_f32, L2-resident
  unsafeAtomicAdd(dp + 1, v.y);
  unsafeAtomicAdd(dp + 2, v.z);
  unsafeAtomicAdd(dp + 3, v.w);
}

// ------------------------------------------------------------- WMMA MLP
// one wave32 handles a 16-node tile.
// GEMM1: [16 x 144] @ [144 x 16] via 36x V_WMMA_F32_16X16X4_F32 (fp32, exact)
// GEMM2: [16 x 16]  @ [16 x 48]  via 12x WMMA (3 output tiles of 16)
__global__ void __launch_bounds__(256)
mlp_softmax_kernel(const float* __restrict__ x,
                   const float* __restrict__ xpos,
                   const float* __restrict__ xneg,
                   const float* __restrict__ W1,   // [144,16] row-major
                   const float* __restrict__ b1,   // [16]
                   const float* __restrict__ W2,   // [16,48] row-major
                   const float* __restrict__ b2,   // [48]
                   float* __restrict__ out,        // [N,48]
                   int nTiles) {
  __shared__ float hsm[8][16 * HID];   // per-wave hidden tile, C-layout -> A-layout bounce

  const int lane = threadIdx.x & 31;
  const int wave = threadIdx.x >> 5;
  const int tile = blockIdx.x * 8 + wave;
  const bool active = tile < nTiles;

  const int m   = lane & 15;   // M (node-in-tile) for A; N column for B/C/D
  const int hi  = lane >> 4;   // half-wave selects K pair in A/B frags, +8 rows in C
  const int off = hi * 2;

  v8f acc = {0.f, 0.f, 0.f, 0.f, 0.f, 0.f, 0.f, 0.f};

  if (active) {
    const long long row = (long long)(tile * 16 + m) * KFEAT;
    const float* seg[3] = { x + row, xpos + row, xneg + row };
    #pragma unroll
    for (int s = 0; s < 3; ++s) {
      const float* p = seg[s];
      const float* w = W1 + (s * KFEAT) * HID;
      #pragma unroll
      for (int c = 0; c < KFEAT / 4; ++c) {
        const int k = 4 * c + off;
        v2f a; a.x = p[k];              a.y = p[k + 1];              // A: 16x4 f32 frag
        v2f b; b.x = w[k * HID + m];    b.y = w[(k + 1) * HID + m];  // B: 4x16 f32 frag
        acc = __builtin_amdgcn_wmma_f32_16x16x4_f32(
            false, a, false, b, (short)0, acc, false, false);
      }
    }
    const float bb = b1[m];
    #pragma unroll
    for (int r = 0; r < 8; ++r) {
      const float h = tanhf(acc[r] + bb);
      hsm[wave][(r + 8 * hi) * HID + m] = h;   // C-layout element (M=r+8*hi, N=m)
    }
  }
  __syncthreads();

  if (active) {
    v8f c0 = {0.f,0.f,0.f,0.f,0.f,0.f,0.f,0.f};
    v8f c1 = c0, c2 = c0;
    const float* hrow = &hsm[wave][m * HID];   // node m's 16 hidden values
    #pragma unroll
    for (int c = 0; c < HID / 4; ++c) {
      const int k = 4 * c + off;
      v2f a; a.x = hrow[k]; a.y = hrow[k + 1];
      const float* w0 = W2 + k * KFEAT;
      const float* w1 = W2 + (k + 1) * KFEAT;
      v2f b;
      b.x = w0[m];       b.y = w1[m];
      c0 = __builtin_amdgcn_wmma_f32_16x16x4_f32(false, a, false, b, (short)0, c0, false, false);
      b.x = w0[16 + m];  b.y = w1[16 + m];
      c1 = __builtin_amdgcn_wmma_f32_16x16x4_f32(false, a, false, b, (short)0, c1, false, false);
      b.x = w0[32 + m];  b.y = w1[32 + m];
      c2 = __builtin_amdgcn_wmma_f32_16x16x4_f32(false, a, false, b, (short)0, c2, false, false);
    }
    const float bias0 = b2[m], bias1 = b2[16 + m], bias2 = b2[32 + m];
    #pragma unroll
    for (int r = 0; r < 8; ++r) {
      float v0 = c0[r] + bias0;
      float v1 = c1[r] + bias1;
      float v2 = c2[r] + bias2;
      // row of node (r + 8*hi) lives across the 16 lanes of this half-wave:
      // butterfly xor 1,2,4,8 stays inside the half -> per-node reduction.
      float mx = fmaxf(v0, fmaxf(v1, v2));
      #pragma unroll
      for (int d = 1; d <= 8; d <<= 1) mx = fmaxf(mx, __shfl_xor(mx, d, 32));
      float e0 = __expf(v0 - mx), e1 = __expf(v1 - mx), e2 = __expf(v2 - mx);
      float sm = e0 + e1 + e2;
      #pragma unroll
      for (int d = 1; d <= 8; d <<= 1) sm += __shfl_xor(sm, d, 32);
      const float inv = 1.f / sm;
      float* o = out + (long long)(tile * 16 + r + 8 * hi) * KFEAT + m;
      o[0]  = e0 * inv;   // coalesced: 16 consecutive floats per half-wave
      o[16] = e1 * inv;
      o[32] = e2 * inv;
    }
  }
}

// ---------------------------------------------------------------- launcher
extern "C" void kernel_launch(void* const* d_in, const int* in_sizes, int n_in,
                              void* d_out, int out_size, void* d_ws, size_t ws_size,
                              hipStream_t stream) {
  const float*     x   = (const float*)d_in[0];
  const long long* pos = (const long long*)d_in[1];
  const long long* neg = (const long long*)d_in[2];
  const float*     W1  = (const float*)d_in[3];
  const float*     b1  = (const float*)d_in[4];
  const float*     W2  = (const float*)d_in[5];
  const float*     b2  = (const float*)d_in[6];
  float* outp = (float*)d_out;

  const int nNodes = in_sizes[0] / KFEAT;       // 100000
  const int E      = in_sizes[1] / 2;           // 1600000

  float* xpos = (float*)d_ws;
  float* xneg = xpos + (size_t)nNodes * KFEAT;

  // 1) zero accumulators (2 * N * K floats, multiple of 4)
  {
    int n4 = (2 * nNodes * KFEAT) / 4;
    int blocks = (n4 + 255) / 256;
    zero_f4_kernel<<<blocks, 256, 0, stream>>>((float4*)d_ws, n4);
  }
  // 2) scatter-add message passing (L2-resident gathers + FP32 HW atomics)
  {
    int total  = E * (KFEAT / 4);
    int blocks = (total + 255) / 256;
    scatter_add_kernel<<<blocks, 256, 0, stream>>>(x, pos, xpos, E);
    scatter_add_kernel<<<blocks, 256, 0, stream>>>(x, neg, xneg, E);
  }
  // 3) fused WMMA MLP + softmax, one wave per 16-node tile
  {
    int nTiles = (nNodes + 15) / 16;            // 6250 (exact)
    int blocks = (nTiles + 7) / 8;
    mlp_softmax_kernel<<<blocks, 256, 0, stream>>>(x, xpos, xneg, W1, b1, W2, b2,
                                                   outp, nTiles);
  }
}